// RGCN_76098230550994
// MI455X (gfx1250) — compile-verified
//
#include <hip/hip_runtime.h>

// RGCN (2 layers) for MI455X / gfx1250.
//
// Phase 1 (dominant, HBM/atomic bound): segmented mean of neighbor features
// per (target, relation) -> agg[N*R, 128] via f32 global atomics. Counts are
// edge-list-only, computed once, shared by both layers.
// Phase 2: fused GEMM  out = relu?( [agg | feat] @ [W_flat ; root] + bias )
// with V_WMMA_F32_16X16X4_F32 (exact f32 path, matches reference precision).
//   - each wave register-blocks 2 M-tiles (32 rows)
//   - B is staged per-64-K-row chunk in LDS, transposed with stride 68 so a
//     ds_load_b64 fragment read hits 64 distinct banks (conflict-free), and
//     all 8 waves of the block share it (8x cut in global B traffic).

#define NUM_RELS 8

typedef __attribute__((ext_vector_type(2))) float v2f;
typedef __attribute__((ext_vector_type(8))) float v8f;

// ---------------------------------------------------------------- counts ---
__global__ __launch_bounds__(256)
void rgcn_count_kernel(const int* __restrict__ tgt, const int* __restrict__ etype,
                       float* __restrict__ cnt, int E) {
    int e = blockIdx.x * blockDim.x + threadIdx.x;
    if (e < E) {
        int seg = tgt[e] * NUM_RELS + etype[e];
        atomicAdd(&cnt[seg], 1.0f);
    }
}

// ----------------------------------------------------------- scatter-add ---
// One wave (32 lanes) per edge; each lane moves a float4 (128 floats/row).
__global__ __launch_bounds__(256)
void rgcn_scatter_kernel(const float* __restrict__ feat,   // [N, 128]
                         const int* __restrict__ src,
                         const int* __restrict__ tgt,
                         const int* __restrict__ etype,
                         float* __restrict__ agg,          // [N*R, 128]
                         int E) {
    int t = blockIdx.x * blockDim.x + threadIdx.x;
    int e    = t >> 5;
    int lane = t & 31;
    if (e >= E) return;
    int s   = src[e];
    int seg = tgt[e] * NUM_RELS + etype[e];
    const float4 v = *(const float4*)(feat + (size_t)s * 128 + lane * 4);
    float* dst = agg + (size_t)seg * 128 + lane * 4;
    atomicAdd(dst + 0, v.x);
    atomicAdd(dst + 1, v.y);
    atomicAdd(dst + 2, v.z);
    atomicAdd(dst + 3, v.w);
}

// ------------------------------------------------------------- normalize ---
__global__ __launch_bounds__(256)
void rgcn_normalize_kernel(float* __restrict__ agg, const float* __restrict__ cnt,
                           int rows) {
    int t = blockIdx.x * blockDim.x + threadIdx.x;
    int row  = t >> 5;
    int lane = t & 31;
    if (row >= rows) return;
    float inv = 1.0f / fmaxf(cnt[row], 1.0f);
    float4* p = (float4*)(agg + (size_t)row * 128 + lane * 4);
    float4 v = *p;
    v.x *= inv; v.y *= inv; v.z *= inv; v.w *= inv;
    *p = v;
}

// ------------------------------------------------------------ fused GEMM ---
// out[N, NT*16] = relu?( A1[N,K1] @ B1 + A2[N,K2] @ B2 + bias )
// Block = 256 threads = 8 waves; each wave owns 32 rows (2 M-tiles).
// WMMA f32 16x16x4 fragments (wave32):
//   A: lane L holds A[L&15][k0 + (L>>4)*2 + i], i = 0,1  (contiguous pair)
//   B: lane L holds B[k0 + (L>>4)*2 + i][L&15]
//   D: lane L, vgpr v holds D[v + (L>>4)*8][L&15]
#define KCHUNK 64
#define BSTRIDE 68   // 64 + 4 pad: bank = (4*col + k) % 64 -> conflict-free b64

template<int NT>
__global__ __launch_bounds__(256)
void rgcn_gemm_kernel(const float* __restrict__ A1, int K1,
                      const float* __restrict__ A2, int K2,
                      const float* __restrict__ B1,   // [K1, NT*16] row-major
                      const float* __restrict__ B2,   // [K2, NT*16] row-major
                      const float* __restrict__ bias, // [NT*16]
                      float* __restrict__ out, int N, int do_relu) {
    const int Dout = NT * 16;
    __shared__ float Bs[NT * 16 * BSTRIDE];

    int waveInBlk = threadIdx.x >> 5;
    int lane = threadIdx.x & 31;
    int half = lane >> 4;            // 0 or 1
    int mr   = lane & 15;            // A row / B,C,D column within tile

    int wbase = (blockIdx.x * 8 + waveInBlk) * 32;     // 32 rows per wave
    int arow0 = wbase + mr;       if (arow0 > N - 1) arow0 = N - 1;
    int arow1 = wbase + 16 + mr;  if (arow1 > N - 1) arow1 = N - 1;

    v8f acc0[NT] = {};
    v8f acc1[NT] = {};

    for (int phase = 0; phase < 2; ++phase) {
        const float* A = phase ? A2 : A1;
        const float* B = phase ? B2 : B1;
        int K = phase ? K2 : K1;
        const float* a0p = A + (size_t)arow0 * K + half * 2;
        const float* a1p = A + (size_t)arow1 * K + half * 2;

        for (int kc = 0; kc < K; kc += KCHUNK) {
            __syncthreads();
            // Stage B[kc..kc+64) into LDS transposed: Bs[col][k].
            // Global reads coalesced along columns.
            for (int idx = threadIdx.x; idx < KCHUNK * Dout; idx += 256) {
                int k = idx / Dout;            // Dout is a power of two
                int c = idx - k * Dout;
                Bs[c * BSTRIDE + k] = B[(size_t)(kc + k) * Dout + c];
            }
            __syncthreads();

            for (int kk = 0; kk < KCHUNK; kk += 4) {
                v2f a0 = *(const v2f*)(a0p + kc + kk);
                v2f a1 = *(const v2f*)(a1p + kc + kk);
#pragma unroll
                for (int j = 0; j < NT; ++j) {
                    v2f b = *(const v2f*)&Bs[(j * 16 + mr) * BSTRIDE + kk + half * 2];
                    acc0[j] = __builtin_amdgcn_wmma_f32_16x16x4_f32(
                        false, a0, false, b, (short)0, acc0[j], false, false);
                    acc1[j] = __builtin_amdgcn_wmma_f32_16x16x4_f32(
                        false, a1, false, b, (short)0, acc1[j], false, false);
                }
            }
        }
    }

    // ---- epilogue: bias (+ relu) + guarded stores ----
#pragma unroll
    for (int j = 0; j < NT; ++j) {
        float bj = bias[j * 16 + mr];
#pragma unroll
        for (int v = 0; v < 8; ++v) {
            int r0 = wbase + v + half * 8;
            float val0 = acc0[j][v] + bj;
            if (do_relu) val0 = fmaxf(val0, 0.0f);
            if (r0 < N) out[(size_t)r0 * Dout + j * 16 + mr] = val0;

            int r1 = wbase + 16 + v + half * 8;
            float val1 = acc1[j][v] + bj;
            if (do_relu) val1 = fmaxf(val1, 0.0f);
            if (r1 < N) out[(size_t)r1 * Dout + j * 16 + mr] = val1;
        }
    }
}

// ---------------------------------------------------------------- launch ---
extern "C" void kernel_launch(void* const* d_in, const int* in_sizes, int n_in,
                              void* d_out, int out_size, void* d_ws, size_t ws_size,
                              hipStream_t stream) {
    const float* x     = (const float*)d_in[0];
    const int*   ei    = (const int*)  d_in[1];   // [2, E]
    const int*   etype = (const int*)  d_in[2];
    const float* W1    = (const float*)d_in[3];   // [R,128,128] -> [1024,128]
    const float* root1 = (const float*)d_in[4];   // [128,128]
    const float* b1    = (const float*)d_in[5];
    const float* W2    = (const float*)d_in[6];   // [R,128,64]  -> [1024,64]
    const float* root2 = (const float*)d_in[7];   // [128,64]
    const float* b2    = (const float*)d_in[8];

    const int R   = NUM_RELS;
    const int Din = 128, Dh = 128;
    const int N = in_sizes[0] / Din;
    const int E = in_sizes[1] / 2;
    const int* src = ei;
    const int* tgt = ei + E;

    // Workspace layout: agg [N*R,128] | cnt [N*R] | h [N,128]
    size_t aggBytes = (size_t)N * R * 128 * sizeof(float);
    size_t cntBytes = (size_t)N * R * sizeof(float);
    char*  ws  = (char*)d_ws;
    float* agg = (float*)ws;
    float* cnt = (float*)(ws + aggBytes);
    float* h   = (float*)(ws + aggBytes + cntBytes);
    float* out = (float*)d_out;

    int cblocks = (E + 255) / 256;                        // 1 thread / edge
    int sblocks = (int)(((long long)E * 32 + 255) / 256); // 1 wave / edge
    int rows    = N * R;
    int nblocks = (int)(((long long)rows * 32 + 255) / 256);
    int gblocks = (N + 255) / 256;                        // 256 rows / block

    // ---- counts (shared by both layers) ----
    hipMemsetAsync(cnt, 0, cntBytes, stream);
    rgcn_count_kernel<<<cblocks, 256, 0, stream>>>(tgt, etype, cnt, E);

    // ---- layer 1 ----
    hipMemsetAsync(agg, 0, aggBytes, stream);
    rgcn_scatter_kernel<<<sblocks, 256, 0, stream>>>(x, src, tgt, etype, agg, E);
    rgcn_normalize_kernel<<<nblocks, 256, 0, stream>>>(agg, cnt, rows);
    rgcn_gemm_kernel<8><<<gblocks, 256, 0, stream>>>(
        agg, R * Din, x, Din, W1, root1, b1, h, N, /*relu=*/1);

    // ---- layer 2 ----
    hipMemsetAsync(agg, 0, aggBytes, stream);
    rgcn_scatter_kernel<<<sblocks, 256, 0, stream>>>(h, src, tgt, etype, agg, E);
    rgcn_normalize_kernel<<<nblocks, 256, 0, stream>>>(agg, cnt, rows);
    rgcn_gemm_kernel<4><<<gblocks, 256, 0, stream>>>(
        agg, R * Dh, h, Dh, W2, root2, b2, out, N, /*relu=*/0);
}